// SelfAttentionBlock_83090437308933
// MI455X (gfx1250) — compile-verified
//
#include <hip/hip_runtime.h>

typedef __attribute__((ext_vector_type(2))) float v2f;
typedef __attribute__((ext_vector_type(8))) float v8f;

#define B_ 16
#define S_ 2048
#define D_ 1024
#define H_ 4096
#define SCALE_ 0.03125f  // 1/sqrt(1024)

static __device__ __forceinline__ v8f wmma_f32(v2f a, v2f b, v8f c) {
  // D = A(16x4 f32) * B(4x16 f32) + C(16x16 f32)
  return __builtin_amdgcn_wmma_f32_16x16x4_f32(false, a, false, b, (short)0, c,
                                               false, false);
}

// ---------------------------------------------------------------------------
// K1: QKV projections.  C[M=32768, N=1024] = X @ W + b, for W in {Wq,Wk,Wv}.
// One wave computes a 16(M) x 64(N) tile with 4 independent accumulators.
// ---------------------------------------------------------------------------
__global__ __launch_bounds__(128) void qkv_gemm_kernel(
    const float* __restrict__ X,
    const float* __restrict__ Wq, const float* __restrict__ bq,
    const float* __restrict__ Wk, const float* __restrict__ bk,
    const float* __restrict__ Wv, const float* __restrict__ bv,
    float* __restrict__ Q, float* __restrict__ K, float* __restrict__ V) {
  const int lane = threadIdx.x & 31;
  const int wave = threadIdx.x >> 5;
  const int row = lane & 15;
  const int half = lane >> 4;

  const float* W;
  const float* bias;
  float* out;
  if (blockIdx.z == 0) { W = Wq; bias = bq; out = Q; }
  else if (blockIdx.z == 1) { W = Wk; bias = bk; out = K; }
  else { W = Wv; bias = bv; out = V; }

  const int mbase = blockIdx.x * 16;
  const int nbase = blockIdx.y * 256 + wave * 64;

  v8f acc0 = {}, acc1 = {}, acc2 = {}, acc3 = {};
  const float* arow = X + (size_t)(mbase + row) * D_ + half * 2;
  const float* wcol = W + nbase + row;

  for (int kk = 0; kk < D_; kk += 4) {
    v2f a = *(const v2f*)(arow + kk);
    const float* w0 = wcol + (size_t)(kk + half * 2) * D_;
    v2f fb0, fb1, fb2, fb3;
    fb0.x = w0[0];  fb0.y = w0[D_];
    fb1.x = w0[16]; fb1.y = w0[D_ + 16];
    fb2.x = w0[32]; fb2.y = w0[D_ + 32];
    fb3.x = w0[48]; fb3.y = w0[D_ + 48];
    acc0 = wmma_f32(a, fb0, acc0);
    acc1 = wmma_f32(a, fb1, acc1);
    acc2 = wmma_f32(a, fb2, acc2);
    acc3 = wmma_f32(a, fb3, acc3);
  }
#pragma unroll
  for (int v = 0; v < 8; ++v) {
    const int m = mbase + half * 8 + v;
    float* o = out + (size_t)m * D_ + nbase + row;
    o[0]  = acc0[v] + bias[nbase + row];
    o[16] = acc1[v] + bias[nbase + 16 + row];
    o[32] = acc2[v] + bias[nbase + 32 + row];
    o[48] = acc3[v] + bias[nbase + 48 + row];
  }
}

// ---------------------------------------------------------------------------
// K2: Pass 1 — per 16-query tile, online softmax stats (row max, row sumexp)
// over all keys.  Score tiles via WMMA f32 16x16x4 with 4 accumulator chains.
// ---------------------------------------------------------------------------
__global__ __launch_bounds__(32) void attn_rowstats_kernel(
    const float* __restrict__ Q, const float* __restrict__ K,
    const int* __restrict__ mask, float* __restrict__ rowmax,
    float* __restrict__ rowsum) {
  const int lane = threadIdx.x & 31;
  const int row = lane & 15;
  const int half = lane >> 4;
  const int b = blockIdx.y;
  const int qbase = blockIdx.x * 16;

  const float* qrow = Q + ((size_t)(b * S_ + qbase + row)) * D_ + half * 2;

  float m[8], s[8];
#pragma unroll
  for (int v = 0; v < 8; ++v) { m[v] = -3.0e38f; s[v] = 0.0f; }

  for (int kt = 0; kt < S_; kt += 16) {
    const float* krow = K + ((size_t)(b * S_ + kt + row)) * D_ + half * 2;
    v8f c0 = {}, c1 = {}, c2 = {}, c3 = {};
    for (int d = 0; d < D_; d += 16) {
      c0 = wmma_f32(*(const v2f*)(qrow + d),      *(const v2f*)(krow + d),      c0);
      c1 = wmma_f32(*(const v2f*)(qrow + d + 4),  *(const v2f*)(krow + d + 4),  c1);
      c2 = wmma_f32(*(const v2f*)(qrow + d + 8),  *(const v2f*)(krow + d + 8),  c2);
      c3 = wmma_f32(*(const v2f*)(qrow + d + 12), *(const v2f*)(krow + d + 12), c3);
    }
    const float colmask = (mask[b * S_ + kt + row] == 0) ? -3.0e38f : 0.0f;
#pragma unroll
    for (int v = 0; v < 8; ++v) {
      float x = (c0[v] + c1[v] + c2[v] + c3[v]) * SCALE_;
      x = (colmask < 0.0f) ? -3.0e38f : x;
      float t = x;
      t = fmaxf(t, __shfl_xor(t, 1));
      t = fmaxf(t, __shfl_xor(t, 2));
      t = fmaxf(t, __shfl_xor(t, 4));
      t = fmaxf(t, __shfl_xor(t, 8));
      const float nm = fmaxf(m[v], t);
      float p = __expf(x - nm);
      p += __shfl_xor(p, 1);
      p += __shfl_xor(p, 2);
      p += __shfl_xor(p, 4);
      p += __shfl_xor(p, 8);
      s[v] = s[v] * __expf(m[v] - nm) + p;
      m[v] = nm;
    }
  }
  if (row == 0) {
#pragma unroll
    for (int v = 0; v < 8; ++v) {
      const int r = b * S_ + qbase + half * 8 + v;
      rowmax[r] = m[v];
      rowsum[r] = s[v];
    }
  }
}

// ---------------------------------------------------------------------------
// K3: Pass 2 — per 16-key tile, recompute scores and accumulate column sums
// of softmax weights:  colsum[b,k] = sum_q exp(s - m_q)/Z_q.
// ---------------------------------------------------------------------------
__global__ __launch_bounds__(32) void attn_colsum_kernel(
    const float* __restrict__ Q, const float* __restrict__ K,
    const int* __restrict__ mask, const float* __restrict__ rowmax,
    const float* __restrict__ rowsum, float* __restrict__ colsum) {
  const int lane = threadIdx.x & 31;
  const int row = lane & 15;
  const int half = lane >> 4;
  const int b = blockIdx.y;
  const int ktbase = blockIdx.x * 16;

  const float* krow = K + ((size_t)(b * S_ + ktbase + row)) * D_ + half * 2;
  const float colmask = (mask[b * S_ + ktbase + row] == 0) ? -3.0e38f : 0.0f;

  float acc = 0.0f;
  for (int qt = 0; qt < S_; qt += 16) {
    const float* qrow = Q + ((size_t)(b * S_ + qt + row)) * D_ + half * 2;
    v8f c0 = {}, c1 = {}, c2 = {}, c3 = {};
    for (int d = 0; d < D_; d += 16) {
      c0 = wmma_f32(*(const v2f*)(qrow + d),      *(const v2f*)(krow + d),      c0);
      c1 = wmma_f32(*(const v2f*)(qrow + d + 4),  *(const v2f*)(krow + d + 4),  c1);
      c2 = wmma_f32(*(const v2f*)(qrow + d + 8),  *(const v2f*)(krow + d + 8),  c2);
      c3 = wmma_f32(*(const v2f*)(qrow + d + 12), *(const v2f*)(krow + d + 12), c3);
    }
    const float* mr = rowmax + b * S_ + qt + half * 8;
    const float* zr = rowsum + b * S_ + qt + half * 8;
#pragma unroll
    for (int v = 0; v < 8; ++v) {
      float x = (c0[v] + c1[v] + c2[v] + c3[v]) * SCALE_;
      x = (colmask < 0.0f) ? -3.0e38f : x;
      acc += __expf(x - mr[v]) / zr[v];
    }
  }
  acc += __shfl_xor(acc, 16);
  if (lane < 16) colsum[b * S_ + ktbase + lane] = acc;
}

// ---------------------------------------------------------------------------
// K4: pooled[b,d] = (1/S) * sum_k colsum[b,k] * V[b,k,d]
// ---------------------------------------------------------------------------
__global__ __launch_bounds__(256) void pooled_kernel(
    const float* __restrict__ V, const float* __restrict__ colsum,
    float* __restrict__ pooled) {
  const int b = blockIdx.y;
  const int d = blockIdx.x * 256 + threadIdx.x;
  const float* vb = V + (size_t)b * S_ * D_ + d;
  const float* cs = colsum + b * S_;
  float acc = 0.0f;
  for (int k = 0; k < S_; ++k) acc += cs[k] * vb[(size_t)k * D_];
  pooled[b * D_ + d] = acc * (1.0f / S_);
}

// ---------------------------------------------------------------------------
// K5/K6: tiny MLP.  h = relu(pooled@W1 + b1);  out = tanh(h@W2 + b2)
// ---------------------------------------------------------------------------
__global__ __launch_bounds__(256) void mlp1_kernel(
    const float* __restrict__ pooled, const float* __restrict__ W1,
    const float* __restrict__ b1, float* __restrict__ h) {
  const int b = blockIdx.y;
  const int j = blockIdx.x * 256 + threadIdx.x;
  const float* p = pooled + b * D_;
  float acc = b1[j];
  for (int d = 0; d < D_; ++d) acc += p[d] * W1[(size_t)d * H_ + j];
  h[b * H_ + j] = fmaxf(acc, 0.0f);
}

__global__ __launch_bounds__(256) void mlp2_kernel(
    const float* __restrict__ h, const float* __restrict__ W2,
    const float* __restrict__ b2, float* __restrict__ out) {
  const int b = blockIdx.y;
  const int d = blockIdx.x * 256 + threadIdx.x;
  const float* hb = h + b * H_;
  float acc = b2[d];
  for (int j = 0; j < H_; ++j) acc += hb[j] * W2[(size_t)j * D_ + d];
  out[b * D_ + d] = tanhf(acc);
}

// ---------------------------------------------------------------------------
extern "C" void kernel_launch(void* const* d_in, const int* in_sizes, int n_in,
                              void* d_out, int out_size, void* d_ws,
                              size_t ws_size, hipStream_t stream) {
  const float* X   = (const float*)d_in[0];
  const int*   msk = (const int*)d_in[1];
  const float* Wq  = (const float*)d_in[2];
  const float* bq  = (const float*)d_in[3];
  const float* Wk  = (const float*)d_in[4];
  const float* bk  = (const float*)d_in[5];
  const float* Wv  = (const float*)d_in[6];
  const float* bv  = (const float*)d_in[7];
  const float* W1  = (const float*)d_in[8];
  const float* b1  = (const float*)d_in[9];
  const float* W2  = (const float*)d_in[10];
  const float* b2  = (const float*)d_in[11];
  float* out = (float*)d_out;

  float* ws = (float*)d_ws;
  const size_t qkv = (size_t)B_ * S_ * D_;
  float* Q      = ws;
  float* K      = Q + qkv;
  float* V      = K + qkv;
  float* rowmax = V + qkv;
  float* rowsum = rowmax + (size_t)B_ * S_;
  float* colsum = rowsum + (size_t)B_ * S_;
  float* pooled = colsum + (size_t)B_ * S_;
  float* h      = pooled + (size_t)B_ * D_;

  qkv_gemm_kernel<<<dim3((B_ * S_) / 16, D_ / 256, 3), 128, 0, stream>>>(
      X, Wq, bq, Wk, bk, Wv, bv, Q, K, V);
  attn_rowstats_kernel<<<dim3(S_ / 16, B_), 32, 0, stream>>>(Q, K, msk, rowmax,
                                                             rowsum);
  attn_colsum_kernel<<<dim3(S_ / 16, B_), 32, 0, stream>>>(Q, K, msk, rowmax,
                                                           rowsum, colsum);
  pooled_kernel<<<dim3(D_ / 256, B_), 256, 0, stream>>>(V, colsum, pooled);
  mlp1_kernel<<<dim3(H_ / 256, B_), 256, 0, stream>>>(pooled, W1, b1, h);
  mlp2_kernel<<<dim3(D_ / 256, B_), 256, 0, stream>>>(h, W2, b2, out);
}